// E84NeuralODECell_10247791968648
// MI455X (gfx1250) — compile-verified
//
#include <hip/hip_runtime.h>

#define T_STEPS 1024
#define BATCH   16
#define DIM     1024
#define NDIM    64
#define N4      256   // 4*n

typedef __attribute__((ext_vector_type(2))) float v2f;
typedef __attribute__((ext_vector_type(8))) float v8f;

// ---------------------------------------------------------------------------
// Phase 1: proj[M=T*B, 256] = X[M, 1024] * W[256, 1024]^T  via V_WMMA_F32_16X16X4_F32
// One workgroup (16 waves) handles one 16-row M-block across all 16 N-tiles,
// so A fragments are shared across waves through WGP$/L2.
// ---------------------------------------------------------------------------
__global__ __launch_bounds__(512) void proj_gemm_wmma(
    const float* __restrict__ X, const float* __restrict__ W,
    float* __restrict__ P) {
  const int tm   = blockIdx.x;            // M-tile index, 0..1023
  const int wave = threadIdx.x >> 5;      // 0..15  -> N-tile index
  const int lane = threadIdx.x & 31;
  const int half = lane >> 4;             // selects K pair (0 -> K0/K1, 1 -> K2/K3)
  const int l16  = lane & 15;

  const float* aptr = X + (size_t)(tm * 16 + l16) * DIM + half * 2;
  const float* bptr = W + (size_t)(wave * 16 + l16) * DIM + half * 2;

  v8f c = {};
#pragma unroll 4
  for (int kk = 0; kk < DIM; kk += 4) {
    v2f a = *(const v2f*)(aptr + kk);
    v2f b = *(const v2f*)(bptr + kk);
    c = __builtin_amdgcn_wmma_f32_16x16x4_f32(false, a, false, b, (short)0, c,
                                              false, false);
  }
  float* prow = P + (size_t)(tm * 16 + half * 8) * N4 + wave * 16 + l16;
#pragma unroll
  for (int v = 0; v < 8; ++v) prow[(size_t)v * N4] = c[v];
}

// ---------------------------------------------------------------------------
// Phase 2: L2-normalize k (cols 0..63) and m (cols 192..255) of each proj row.
// ---------------------------------------------------------------------------
__global__ __launch_bounds__(32) void normalize_km(float* __restrict__ P) {
  const int row  = blockIdx.x;          // 0..16383
  const int lane = threadIdx.x;         // 0..31
  float* p = P + (size_t)row * N4;

  float k0 = p[lane], k1 = p[lane + 32];
  float m0 = p[192 + lane], m1 = p[192 + 32 + lane];
  float sk = k0 * k0 + k1 * k1;
  float sm = m0 * m0 + m1 * m1;
#pragma unroll
  for (int off = 1; off < 32; off <<= 1) {
    sk += __shfl_xor(sk, off, 32);
    sm += __shfl_xor(sm, off, 32);
  }
  float rk = __builtin_amdgcn_rcpf(sqrtf(sk) + 1e-6f);
  float rm = __builtin_amdgcn_rcpf(sqrtf(sm) + 1e-6f);
  p[lane]            = k0 * rk;
  p[lane + 32]       = k1 * rk;
  p[192 + lane]      = m0 * rm;
  p[192 + 32 + lane] = m1 * rm;
}

// ---------------------------------------------------------------------------
// Phase 3: sequential RK4 scan, algebraically reduced.
// Within one t-step every RK4 eval point is per-row affine in the base state:
//   S_hat[i,:] = a[i]*S[i,:] + b[i]*k ,  G_hat[i,:] = aG[i]*G[i,:] + bG[i]*m.
// All 16 dynamics evals reduce to scalar recurrences on the row dots
// (S.k, S.m, G.k, G.m) with k.k, k.m, m.m; the matrices are only read for
// 4 matvecs at step start and written once by a rank-1 reconstruction.
// 256 threads/block: thread(row = tid>>2, cg = tid&3) owns 16 columns.
// ---------------------------------------------------------------------------
__device__ __forceinline__ float rowreduce4(float x) {
  x += __shfl_xor(x, 1, 32);
  x += __shfl_xor(x, 2, 32);
  return x;
}

__device__ __forceinline__ float sigmoidf(float x) {
  // v_exp_f32 + v_rcp_f32: short dependent chain, ~1 ulp (plenty for sigmoid)
  return __builtin_amdgcn_rcpf(1.f + __expf(-x));
}

// One dynamics eval in coefficient space.
__device__ __forceinline__ void scalar_eval(
    float aS, float bS, float aG, float bG,
    float sk, float sm, float gk, float gm,
    float kk, float km, float mm, float v_row,
    float& pS, float& qS, float& pG, float& qG) {
  float hsk = aS * sk + bS * kk;   // S_hat . k
  float hsm = aS * sm + bS * km;   // S_hat . m
  float hgk = aG * gk + bG * km;   // G_hat . k
  float hgm = aG * gm + bG * mm;   // G_hat . m
  float dS_ = v_row - hsk;         // delta_S row value
  float dG_ = dS_ - hgm;           // delta_G row value
  float gS = sigmoidf(hgk);        // gate_S = sigmoid(G.k)
  float gG = sigmoidf(hsm);        // gate_G = sigmoid(S.m)
  float gs1 = gS - 1.f, gg1 = gG - 1.f;
  pS = gs1 * aS;  qS = gs1 * bS + dS_;   // dS = pS*S + qS (x) k
  pG = gg1 * aG;  qG = gg1 * bG + dG_;   // dG = pG*G + qG (x) m
}

__global__ __launch_bounds__(256) void ode_scan(
    const float* __restrict__ P, const float* __restrict__ S0,
    const float* __restrict__ G0, float* __restrict__ out,
    float* __restrict__ Sout, float* __restrict__ Gout) {
  const int b     = blockIdx.x;       // batch 0..15
  const int tid   = threadIdx.x;      // 0..255
  const int row   = tid >> 2;         // 0..63
  const int cg    = tid & 3;          // 0..3
  const int cbase = cg * 16;

  __shared__ float lds[N4];

  const size_t sg_off = (size_t)b * NDIM * NDIM + (size_t)row * NDIM + cbase;
  float S[16], G[16];
#pragma unroll
  for (int i = 0; i < 16; ++i) {
    S[i] = S0[sg_off + i];
    G[i] = G0[sg_off + i];
  }

  const float dt  = 0.25f;          // DT = 1/N_STEPS
  const float hdt = 0.5f * dt;
  const float dt6 = dt / 6.0f;

  // software prefetch of the first proj row
  float pr = P[(size_t)b * N4 + tid];

  for (int t = 0; t < T_STEPS; ++t) {
    __syncthreads();
    lds[tid] = pr;
    __syncthreads();

    float kc[16], mc[16], qc[16];
#pragma unroll
    for (int i = 0; i < 16; ++i) {
      kc[i] = lds[cbase + i];          // k = cols 0..63
      mc[i] = lds[192 + cbase + i];    // m = cols 192..255
      qc[i] = lds[128 + cbase + i];    // q = cols 128..191
    }
    const float v_row = lds[64 + row]; // v = cols 64..127

    // prefetch next step's row; wait lands at next iteration's LDS store,
    // fully hidden behind this step's compute
    if (t + 1 < T_STEPS) pr = P[((size_t)(t + 1) * BATCH + b) * N4 + tid];

    // --- dot scalars for this step (4-lane row groups span all 64 cols) ---
    float pkk = 0.f, pkm = 0.f, pmm = 0.f;
    float psk = 0.f, psm = 0.f, pgk = 0.f, pgm = 0.f;
#pragma unroll
    for (int i = 0; i < 16; ++i) {
      pkk += kc[i] * kc[i];
      pkm += kc[i] * mc[i];
      pmm += mc[i] * mc[i];
      psk += S[i] * kc[i];
      psm += S[i] * mc[i];
      pgk += G[i] * kc[i];
      pgm += G[i] * mc[i];
    }
    const float kk = rowreduce4(pkk);
    const float km = rowreduce4(pkm);
    const float mm = rowreduce4(pmm);
    float sk = rowreduce4(psk);
    float sm = rowreduce4(psm);
    float gk = rowreduce4(pgk);
    float gm = rowreduce4(pgm);

    // totals over the 4 RK4 steps: S_final = AtS*S + BtS (x) k
    float AtS = 1.f, BtS = 0.f, AtG = 1.f, BtG = 0.f;

#pragma unroll
    for (int step = 0; step < 4; ++step) {
      float p1S, q1S, p1G, q1G, p2S, q2S, p2G, q2G;
      float p3S, q3S, p3G, q3G, p4S, q4S, p4G, q4G;
      float aS, bS, aG, bG;

      // k1 at base
      scalar_eval(1.f, 0.f, 1.f, 0.f, sk, sm, gk, gm, kk, km, mm, v_row,
                  p1S, q1S, p1G, q1G);
      aS = 1.f + hdt * p1S; bS = hdt * q1S;
      aG = 1.f + hdt * p1G; bG = hdt * q1G;
      // k2
      scalar_eval(aS, bS, aG, bG, sk, sm, gk, gm, kk, km, mm, v_row,
                  p2S, q2S, p2G, q2G);
      aS = 1.f + hdt * p2S; bS = hdt * q2S;
      aG = 1.f + hdt * p2G; bG = hdt * q2G;
      // k3
      scalar_eval(aS, bS, aG, bG, sk, sm, gk, gm, kk, km, mm, v_row,
                  p3S, q3S, p3G, q3G);
      aS = 1.f + dt * p3S; bS = dt * q3S;
      aG = 1.f + dt * p3G; bG = dt * q3G;
      // k4
      scalar_eval(aS, bS, aG, bG, sk, sm, gk, gm, kk, km, mm, v_row,
                  p4S, q4S, p4G, q4G);

      const float A  = 1.f + dt6 * (p1S + 2.f * p2S + 2.f * p3S + p4S);
      const float Bc = dt6 * (q1S + 2.f * q2S + 2.f * q3S + q4S);
      const float Ag = 1.f + dt6 * (p1G + 2.f * p2G + 2.f * p3G + p4G);
      const float Bg = dt6 * (q1G + 2.f * q2G + 2.f * q3G + q4G);

      // propagate base dots to the next RK4 step
      sk = A * sk + Bc * kk;   sm = A * sm + Bc * km;
      gk = Ag * gk + Bg * km;  gm = Ag * gm + Bg * mm;

      // accumulate rank-1 totals (order matters: B before A)
      BtS = A * BtS + Bc;   AtS = A * AtS;
      BtG = Ag * BtG + Bg;  AtG = Ag * AtG;
    }

    // --- one rank-1 reconstruction of the matrices per t-step ---
#pragma unroll
    for (int i = 0; i < 16; ++i) {
      S[i] = AtS * S[i] + BtS * kc[i];
      G[i] = AtG * G[i] + BtG * mc[i];
    }

    // output: Sq = S.q ; out = Sq * silu(Sq) = Sq^2 * sigmoid(Sq)
    float pq = 0.f;
#pragma unroll
    for (int i = 0; i < 16; ++i) pq += S[i] * qc[i];
    pq = rowreduce4(pq);
    if (cg == 0) {
      out[((size_t)t * BATCH + b) * NDIM + row] = pq * pq * sigmoidf(pq);
    }
  }

#pragma unroll
  for (int i = 0; i < 16; ++i) {
    Sout[sg_off + i] = S[i];
    Gout[sg_off + i] = G[i];
  }
}

// ---------------------------------------------------------------------------
extern "C" void kernel_launch(void* const* d_in, const int* in_sizes, int n_in,
                              void* d_out, int out_size, void* d_ws, size_t ws_size,
                              hipStream_t stream) {
  (void)in_sizes; (void)n_in; (void)out_size; (void)ws_size;
  const float* x  = (const float*)d_in[0];   // [T, B, D]
  const float* W  = (const float*)d_in[1];   // [4n, D]
  const float* S0 = (const float*)d_in[2];   // [B, n, n]
  const float* G0 = (const float*)d_in[3];   // [B, n, n]

  float* out  = (float*)d_out;                           // [T, B, n]
  float* Sout = out + (size_t)T_STEPS * BATCH * NDIM;    // [B, n, n]
  float* Gout = Sout + (size_t)BATCH * NDIM * NDIM;      // [B, n, n]

  float* P = (float*)d_ws;  // proj [T*B, 256] = 16 MB scratch

  proj_gemm_wmma<<<dim3((T_STEPS * BATCH) / 16, 1, 1), 512, 0, stream>>>(x, W, P);
  normalize_km<<<T_STEPS * BATCH, 32, 0, stream>>>(P);
  ode_scan<<<BATCH, 256, 0, stream>>>(P, S0, G0, out, Sout, Gout);
}